// Model_18726057411280
// MI455X (gfx1250) — compile-verified
//
#include <hip/hip_runtime.h>
#include <stdint.h>

// Flash-attention (fp32) for [B=2,H=16,S=2048,D=128] on gfx1250 (MI455X).
// HBM-bound (~134MB vs ~69 GFLOP) -> single pass over K/V, fp32 WMMA
// (V_WMMA_F32_16X16X4_F32), K/V tiles DMA'd into double-buffered LDS with
// TENSOR_LOAD_TO_LDS (TDM) overlapped with compute, TENSORcnt-synchronized.

typedef __attribute__((ext_vector_type(2))) float v2f;
typedef __attribute__((ext_vector_type(4))) float v4f;
typedef __attribute__((ext_vector_type(8))) float v8f;
typedef __attribute__((ext_vector_type(4))) unsigned int v4u;
typedef __attribute__((ext_vector_type(8))) int v8i;
typedef __attribute__((ext_vector_type(4))) int v4i;

#define SEQ 2048
#define DIM 128
#define NTILES (SEQ / 16)
#define QK_SCALE 0.088388f

#define LK_STRIDE 132  // 128 + 4 pad dwords (TDM pad) -> conflict-free b64 reads
#define LP_STRIDE 18   // 16 + 2 pad

#if __has_builtin(__builtin_amdgcn_tensor_load_to_lds) && \
    __has_builtin(__builtin_amdgcn_s_wait_tensorcnt)
#define HAVE_TDM 1
#else
#define HAVE_TDM 0
#endif

__global__ __launch_bounds__(256) void fa_fwd_f32_wmma(
    const float* __restrict__ Q, const float* __restrict__ K,
    const float* __restrict__ V, float* __restrict__ O) {
  __shared__ float ldsK[2 * 16 * LK_STRIDE];   // K tiles, padded row-major
  __shared__ float ldsV[2 * 16 * DIM];         // V tiles, row-major [key][d]
  __shared__ float ldsP[8 * 16 * LP_STRIDE];   // per-wave P staging [m][k]

  const int bh   = blockIdx.y;            // 0..31  (b*16+h)
  const int lane = threadIdx.x & 31;
  const int wv   = threadIdx.x >> 5;      // wave 0..7
  const int hf   = lane >> 4;             // half-wave 0/1
  const int lm   = lane & 15;             // lane-in-half

  const size_t base = (size_t)bh * SEQ * DIM;
  const int q0 = blockIdx.x * 128 + wv * 16;   // this wave's 16 query rows

  // --- Q fragment in WMMA A-layout (lane=M, vgpr=K pair), pre-scaled ---
  const float* qrow = Q + base + (size_t)(q0 + lm) * DIM;
  v2f a[32];
#pragma unroll
  for (int kc = 0; kc < 32; ++kc) {
    v2f t = *(const v2f*)(qrow + kc * 4 + 2 * hf);
    a[kc] = t * QK_SCALE;
  }

  v8f zero8 = {};
  v8f o[8];
#pragma unroll
  for (int nt = 0; nt < 8; ++nt) o[nt] = zero8;
  float mrow[8], lrow[8];
#pragma unroll
  for (int r = 0; r < 8; ++r) { mrow[r] = -__builtin_inff(); lrow[r] = 0.0f; }

  float* ldsPw = ldsP + wv * 16 * LP_STRIDE;

#if HAVE_TDM
  // Each wave DMAs its 2-row slice of the K tile (TDM-padded to stride 132)
  // and of the V tile (dense) via TENSOR_LOAD_TO_LDS. 2 TDM ops / wave / tile.
  const uint32_t ldsKbase = (uint32_t)(uintptr_t)&ldsK[0];
  const uint32_t ldsVbase = (uint32_t)(uintptr_t)&ldsV[0];
  const v4i zero4 = {0, 0, 0, 0};
  const v8i zero8i = {0, 0, 0, 0, 0, 0, 0, 0};
  // group1 word0: data_size=4B (bits 17:16 = 2); K adds pad: enable bit20,
  // interval=6 (128 dwords, bits 24:22), amount=3 (4 dwords, bits 31:25).
  const int g1w0_k = (int)(0x20000u | (1u << 20) | (6u << 22) | (3u << 25));
  const int g1w0_v = (int)0x20000u;

  auto issue_tile = [&](int tile, int bufsel) {
    const size_t grow = (size_t)(tile * 16 + 2 * wv) * DIM;
    const uint64_t gk = (uint64_t)(uintptr_t)(K + base + grow);
    const uint64_t gv = (uint64_t)(uintptr_t)(V + base + grow);
    const uint32_t lk = ldsKbase + (uint32_t)(bufsel * 16 + 2 * wv) * LK_STRIDE * 4u;
    const uint32_t lv = ldsVbase + (uint32_t)(bufsel * 16 + 2 * wv) * DIM * 4u;

    v4u g0k = {1u, lk, (uint32_t)gk, (uint32_t)(gk >> 32) | (2u << 30)};
    v8i g1k = {g1w0_k,
               (int)(128u << 16),  // tensor_dim0 = 128
               (int)(2u << 16),    // tensor_dim1 = 2
               (int)(128u << 16),  // tile_dim0 = 128
               2,                  // tile_dim1 = 2
               128,                // tensor_dim0_stride = 128
               0, 0};
    __builtin_amdgcn_tensor_load_to_lds(g0k, g1k, zero4, zero4, zero8i, 0);

    v4u g0v = {1u, lv, (uint32_t)gv, (uint32_t)(gv >> 32) | (2u << 30)};
    v8i g1v = {g1w0_v,
               (int)(128u << 16), (int)(2u << 16), (int)(128u << 16),
               2, 128, 0, 0};
    __builtin_amdgcn_tensor_load_to_lds(g0v, g1v, zero4, zero4, zero8i, 0);
  };

  issue_tile(0, 0);  // prologue: tile 0 -> buffer 0
#else
  // Fallback: synchronous cooperative staging (thread t: key row t/16,
  // d-block (t%16)*8), K padded to LK_STRIDE, V dense row-major.
  const int stkey = threadIdx.x >> 4;
  const int std_  = (threadIdx.x & 15) * 8;
#endif

  for (int t = 0; t < NTILES; ++t) {
    const int n0 = t * 16;
    const float* kbuf = ldsK + (t & 1) * 16 * LK_STRIDE;
    const float* vbuf = ldsV + (t & 1) * 16 * DIM;

#if HAVE_TDM
    __syncthreads();  // everyone done reading buffer (t+1)&1 (iter t-1)
    if (t + 1 < NTILES) {
      issue_tile(t + 1, (t + 1) & 1);
      __builtin_amdgcn_s_wait_tensorcnt(2);  // tile t done, t+1 in flight
    } else {
      __builtin_amdgcn_s_wait_tensorcnt(0);
    }
    __syncthreads();  // tile t visible to all waves
#else
    __syncthreads();
    {
      const float* kp = K + base + (size_t)(n0 + stkey) * DIM + std_;
      const float* vp = V + base + (size_t)(n0 + stkey) * DIM + std_;
      v4f k0 = *(const v4f*)kp;
      v4f k1 = *(const v4f*)(kp + 4);
      v4f v0 = *(const v4f*)vp;
      v4f v1 = *(const v4f*)(vp + 4);
      float* kw = ldsK + (t & 1) * 16 * LK_STRIDE;
      float* vw = ldsV + (t & 1) * 16 * DIM;
      *(v4f*)&kw[stkey * LK_STRIDE + std_]     = k0;
      *(v4f*)&kw[stkey * LK_STRIDE + std_ + 4] = k1;
      *(v4f*)&vw[stkey * DIM + std_]           = v0;
      *(v4f*)&vw[stkey * DIM + std_ + 4]       = v1;
    }
    __syncthreads();
#endif

    // ---- S = (Q*scale) x K^T : 16x16 tile, K-dim 128 in chunks of 4 ----
    v8f s = zero8;
#pragma unroll
    for (int kc = 0; kc < 32; ++kc) {
      v2f b = *(const v2f*)&kbuf[lm * LK_STRIDE + kc * 4 + 2 * hf];
      s = __builtin_amdgcn_wmma_f32_16x16x4_f32(false, a[kc], false, b,
                                                (short)0, s, false, false);
    }

    // ---- online softmax: row r lives in lanes of this half-wave ----
    float corr[8];
#pragma unroll
    for (int r = 0; r < 8; ++r) {
      float v = s[r];
#pragma unroll
      for (int msk = 1; msk < 16; msk <<= 1)
        v = fmaxf(v, __shfl_xor(v, msk, 32));
      float nm = fmaxf(mrow[r], v);
      corr[r] = __expf(mrow[r] - nm);
      mrow[r] = nm;
      float p = __expf(s[r] - nm);
      s[r] = p;
      float sum = p;
#pragma unroll
      for (int msk = 1; msk < 16; msk <<= 1)
        sum += __shfl_xor(sum, msk, 32);
      lrow[r] = lrow[r] * corr[r] + sum;
    }
#pragma unroll
    for (int nt = 0; nt < 8; ++nt)
#pragma unroll
      for (int r = 0; r < 8; ++r) o[nt][r] *= corr[r];

    // ---- P: C-layout -> A-layout via per-wave LDS round-trip ----
#pragma unroll
    for (int r = 0; r < 8; ++r)
      ldsPw[(r + 8 * hf) * LP_STRIDE + lm] = s[r];
    asm volatile("s_wait_dscnt 0" ::: "memory");  // DS is in-order per wave
    v2f pa[4];
#pragma unroll
    for (int kc = 0; kc < 4; ++kc)
      pa[kc] = *(const v2f*)&ldsPw[lm * LP_STRIDE + kc * 4 + 2 * hf];

    // ---- O += P x V (K-dim 16 in chunks of 4, 8 n-tiles over D) ----
#pragma unroll
    for (int kc = 0; kc < 4; ++kc)
#pragma unroll
      for (int nt = 0; nt < 8; ++nt) {
        const int kk = kc * 4 + 2 * hf;
        v2f bv = {vbuf[kk * DIM + nt * 16 + lm],
                  vbuf[(kk + 1) * DIM + nt * 16 + lm]};
        o[nt] = __builtin_amdgcn_wmma_f32_16x16x4_f32(false, pa[kc], false, bv,
                                                      (short)0, o[nt], false, false);
      }
    (void)n0;
  }

  // ---- normalize and store ----
  float* orow = O + base + (size_t)q0 * DIM;
#pragma unroll
  for (int r = 0; r < 8; ++r) {
    float inv = 1.0f / lrow[r];
#pragma unroll
    for (int nt = 0; nt < 8; ++nt)
      orow[(size_t)(r + 8 * hf) * DIM + nt * 16 + lm] = o[nt][r] * inv;
  }
}

extern "C" void kernel_launch(void* const* d_in, const int* in_sizes, int n_in,
                              void* d_out, int out_size, void* d_ws, size_t ws_size,
                              hipStream_t stream) {
  (void)in_sizes; (void)n_in; (void)out_size; (void)d_ws; (void)ws_size;
  const float* Q = (const float*)d_in[0];   // var
  const float* K = (const float*)d_in[1];   // var_scale
  const float* V = (const float*)d_in[2];   // indices
  float* O = (float*)d_out;

  dim3 grid(SEQ / 128, 2 * 16);  // (query tiles, B*H)
  dim3 block(256);               // 8 wave32
  fa_fwd_f32_wmma<<<grid, block, 0, stream>>>(Q, K, V, O);
}